// DAMSMAttention_85280870629865
// MI455X (gfx1250) — compile-verified
//
#include <hip/hip_runtime.h>
#include <math.h>

// DAMSM attention for MI455X (gfx1250), fp32 WMMA path.
// B=128, N=289 (17x17), T=32, D=1024, GAMMA1=4.

typedef __attribute__((ext_vector_type(2))) float v2f;
typedef __attribute__((ext_vector_type(8))) float v8f;

#define BATCH  128
#define NREG   289
#define TWORDS 32
#define DDIM   1024
#define GAMMA1 4.0f
#define NTILES 19     // ceil(289/16)
#define NPAD   292    // 289 padded to multiple of 4 (keeps v2f LDS reads aligned)

// ---------------------------------------------------------------------------
// Kernel 1: attnT[t][n] = sum_d word[b,t,d] * ctx[b,n,d]  (WMMA f32 16x16x4)
// then softmax over t (32 values) per column n, write to ws as [B, T, N].
// Grid: (19 n-tiles, 128 batches), Block: 64 threads = 2 waves (t-halves).
// ---------------------------------------------------------------------------
__global__ __launch_bounds__(64)
void k1_gemm1_softmaxT(const float* __restrict__ ctx,   // [B, N, D]
                       const float* __restrict__ word,  // [B, T, D]
                       float* __restrict__ attn)        // [B, T, N] (ws)
{
    __shared__ float s[TWORDS][16];

    const int ntile = blockIdx.x;
    const int b     = blockIdx.y;
    const int lane  = threadIdx.x & 31;
    const int wave  = threadIdx.x >> 5;        // t-half: 0 -> t 0..15, 1 -> t 16..31
    const int l16   = lane & 15;
    const int khalf = (lane >> 4) << 1;        // 0 for lanes 0-15, 2 for lanes 16-31

    const int n0 = ntile * 16;
    const int n  = n0 + l16;
    const int ncl = (n < NREG) ? n : (NREG - 1);   // clamp: garbage cols never written

    // A operand (word): lane l16 holds row t = wave*16+l16, VGPR0/1 = K = k0+khalf(+1)
    const float* wrow = word + ((size_t)b * TWORDS + wave * 16 + l16) * DDIM + khalf;
    // B operand (ctx):  lane l16 holds col n, VGPR0/1 = K = k0+khalf(+1)
    const float* crow = ctx + ((size_t)b * NREG + ncl) * DDIM + khalf;

    v8f acc = {};
#pragma unroll 8
    for (int k0 = 0; k0 < DDIM; k0 += 4) {
        v2f a  = *(const v2f*)(wrow + k0);
        v2f bb = *(const v2f*)(crow + k0);
        acc = __builtin_amdgcn_wmma_f32_16x16x4_f32(false, a, false, bb,
                                                    (short)0, acc, false, false);
    }

    // C layout: VGPR v, lanes 0-15 -> M=v; lanes 16-31 -> M=v+8; col = l16
    const int rbase = wave * 16 + ((lane >> 4) << 3);
#pragma unroll
    for (int v = 0; v < 8; ++v)
        s[rbase + v][l16] = acc[v];
    __syncthreads();

    // softmax over t (length 32) per column; threads 0..15 each own a column
    if (threadIdx.x < 16) {
        const int c = threadIdx.x;
        float m = -INFINITY;
#pragma unroll
        for (int t = 0; t < TWORDS; ++t) m = fmaxf(m, s[t][c]);
        float sum = 0.f;
#pragma unroll
        for (int t = 0; t < TWORDS; ++t) { float e = expf(s[t][c] - m); s[t][c] = e; sum += e; }
        const float inv = 1.f / sum;
#pragma unroll
        for (int t = 0; t < TWORDS; ++t) s[t][c] *= inv;
    }
    __syncthreads();

    // write valid columns to ws[b][t][n]
    const int nv = (NREG - n0 < 16) ? (NREG - n0) : 16;
    for (int i = threadIdx.x; i < TWORDS * nv; i += 64) {
        const int t = i / nv;
        const int j = i - t * nv;
        attn[((size_t)b * TWORDS + t) * NREG + n0 + j] = s[t][j];
    }
}

// ---------------------------------------------------------------------------
// Kernel 1b: in-place softmax over n (length 289) of GAMMA1 * x, per (b,t) row.
// Grid: 4096 rows, Block: 128.
// ---------------------------------------------------------------------------
__global__ __launch_bounds__(128)
void k1b_softmaxN(float* __restrict__ attn)   // [B, T, N]
{
    __shared__ float red[128];
    float* p = attn + (size_t)blockIdx.x * NREG;
    const int tid = threadIdx.x;

    float m = -INFINITY;
    for (int i = tid; i < NREG; i += 128) m = fmaxf(m, GAMMA1 * p[i]);
    red[tid] = m; __syncthreads();
    for (int s = 64; s > 0; s >>= 1) {
        if (tid < s) red[tid] = fmaxf(red[tid], red[tid + s]);
        __syncthreads();
    }
    m = red[0];
    __syncthreads();

    float sum = 0.f;
    for (int i = tid; i < NREG; i += 128) {
        float e = expf(GAMMA1 * p[i] - m);
        p[i] = e;
        sum += e;
    }
    red[tid] = sum; __syncthreads();
    for (int s = 64; s > 0; s >>= 1) {
        if (tid < s) red[tid] += red[tid + s];
        __syncthreads();
    }
    const float inv = 1.f / red[0];
    for (int i = tid; i < NREG; i += 128) p[i] *= inv;
}

// ---------------------------------------------------------------------------
// Kernel 2: out[b,d,t] = sum_n ctx[b,n,d] * attn2[b,t,n]   (WMMA f32 16x16x4)
// Grid: (16 d-tiles of 64, 128 batches), Block: 128 = 4 waves (16 d each).
// attn2[b] (32x289) staged in LDS padded to 292 cols (pad reads = 0).
// ---------------------------------------------------------------------------
__global__ __launch_bounds__(128)
void k2_gemm2(const float* __restrict__ ctx,   // [B, N, D]
              const float* __restrict__ attn,  // [B, T, N]
              float* __restrict__ out)         // [B, D, T]
{
    __shared__ float s[TWORDS][NPAD];          // 37376 bytes

    const int b    = blockIdx.y;
    const int d0   = blockIdx.x * 64;
    const int tid  = threadIdx.x;
    const int lane = tid & 31;
    const int wave = tid >> 5;
    const int l16  = lane & 15;
    const int khalf = (lane >> 4) << 1;

    // stage attn2[b] into LDS, zero the 3-column pad
    const float* ap = attn + (size_t)b * TWORDS * NREG;
    for (int i = tid; i < TWORDS * NREG; i += 128) {
        const int t = i / NREG;
        s[t][i - t * NREG] = ap[i];
    }
    for (int i = tid; i < TWORDS * (NPAD - NREG); i += 128) {
        const int t = i / (NPAD - NREG);
        s[t][NREG + (i - t * (NPAD - NREG))] = 0.f;
    }
    __syncthreads();

    // A operand (ctx^T): lane l16 holds row d = d0 + wave*16 + l16,
    //                    VGPR0/1 = K(=n) = k0+khalf(+1); coalesced across lanes.
    const int d = d0 + wave * 16 + l16;
    const float* cbase = ctx + (size_t)b * NREG * DDIM + d;

    v8f acc0 = {}, acc1 = {};
    for (int k0 = 0; k0 < NREG; k0 += 4) {     // last iter k0=288: tail guarded
        const int kA = k0 + khalf;
        v2f a;
        a.x = (kA     < NREG) ? cbase[(size_t)kA * DDIM]       : 0.f;
        a.y = (kA + 1 < NREG) ? cbase[(size_t)(kA + 1) * DDIM] : 0.f;
        // B operand from LDS: lane l16 is col t (or t+16); pad cols are zero
        v2f b0 = *(const v2f*)&s[l16][kA];
        v2f b1 = *(const v2f*)&s[16 + l16][kA];
        acc0 = __builtin_amdgcn_wmma_f32_16x16x4_f32(false, a, false, b0,
                                                     (short)0, acc0, false, false);
        acc1 = __builtin_amdgcn_wmma_f32_16x16x4_f32(false, a, false, b1,
                                                     (short)0, acc1, false, false);
    }

    // store: acc[v] lane<16 -> d-row v, col l16; lane>=16 -> d-row v+8
    const int mrow = wave * 16 + ((lane >> 4) << 3);
    float* op = out + (size_t)b * DDIM * TWORDS;
#pragma unroll
    for (int v = 0; v < 8; ++v) {
        const int dd = d0 + mrow + v;
        op[(size_t)dd * TWORDS + l16]      = acc0[v];
        op[(size_t)dd * TWORDS + 16 + l16] = acc1[v];
    }
}

extern "C" void kernel_launch(void* const* d_in, const int* in_sizes, int n_in,
                              void* d_out, int out_size, void* d_ws, size_t ws_size,
                              hipStream_t stream) {
    const float* ctx  = (const float*)d_in[0];   // image_feature [128,17,17,1024]
    const float* word = (const float*)d_in[1];   // word_vector  [128,32,1024]
    float* out  = (float*)d_out;                 // [128,1024,32]
    float* attn = (float*)d_ws;                  // [128,32,289] fp32 (~4.5 MB)

    k1_gemm1_softmaxT<<<dim3(NTILES, BATCH), 64, 0, stream>>>(ctx, word, attn);
    k1b_softmaxN<<<dim3(BATCH * TWORDS), 128, 0, stream>>>(attn);
    k2_gemm2<<<dim3(DDIM / 64, BATCH), 128, 0, stream>>>(ctx, attn, out);
}